// Softarea_68221260529729
// MI455X (gfx1250) — compile-verified
//
#include <hip/hip_runtime.h>
#include <stdint.h>

// ---------------- problem constants (reference: H=W=256) ----------------
constexpr int Lleaves = 256 * 256;          // 65536 leaves
constexpr int N_NODES = 2 * Lleaves - 1;    // 131071 tree nodes
constexpr int NPOW    = 131072;             // pow2 padding for bitonic sort / arrays
constexpr int NSTAGE  = Lleaves;            // internal-parent slots staged to LDS (256 KB)

typedef __attribute__((ext_vector_type(4))) unsigned int u32x4;
typedef __attribute__((ext_vector_type(8))) int          i32x8;
typedef __attribute__((ext_vector_type(4))) int          i32x4;

__device__ __forceinline__ float sigmoidf(float x) {
    return 1.0f / (1.0f + __expf(-x));
}

// ---------------------------------------------------------------------------
// TDM bulk copy: parent[Lleaves .. Lleaves+65536) -> LDS.
// One descriptor: 16 rows x 4096 elements of 4-byte data, contiguous.
// Issued by wave 0 only (EXEC ignored by tensor ops; descriptor is uniform).
// ---------------------------------------------------------------------------
__device__ __forceinline__ void stage_parents_to_lds(const int* gparent, int* ldsPtr) {
    if (threadIdx.x < 32) {
        uint64_t gaddr = (uint64_t)(uintptr_t)(gparent + Lleaves);
        uint32_t laddr = (uint32_t)(uintptr_t)ldsPtr;   // LDS aperture: low 32 bits = LDS offset
        // D# group0: count=1 | lds_addr | global_addr[56:0] | type=2
        u32x4 g0 = { 1u,
                     laddr,
                     (uint32_t)gaddr,
                     (uint32_t)((gaddr >> 32) & 0x1FFFFFFu) | (2u << 30) };
        // D# group1: data_size=4B; tensor 65536x16; tile 4096x16; dim0 stride 4096
        i32x8 g1;
        g1[0] = 0x00020000;           // workgroup_mask=0, data_size=2 (4B), no pad/iterate
        g1[1] = 0;                    // atomic_barrier=0, tensor_dim0 low16 (65536&0xFFFF)
        g1[2] = (16 << 16) | 1;       // tensor_dim0 hi16 = 1 (=> 65536), tensor_dim1 = 16
        g1[3] = (4096 << 16);         // tile_dim0 = 4096
        g1[4] = 16;                   // tile_dim1 = 16, tile_dim2 = 0
        g1[5] = 4096;                 // tensor_dim0_stride = 4096 (low 32)
        g1[6] = 0;
        g1[7] = 0;
        i32x4 z4 = {0, 0, 0, 0};
#if __has_include(<hip/amd_detail/amd_gfx1250_TDM.h>)
        i32x8 z8 = {0, 0, 0, 0, 0, 0, 0, 0};
        __builtin_amdgcn_tensor_load_to_lds(g0, g1, z4, z4, z8, 0);
#else
        __builtin_amdgcn_tensor_load_to_lds(g0, g1, z4, z4, 0);
#endif
        __builtin_amdgcn_s_wait_tensorcnt(0);
    }
    __syncthreads();
}

// ---------------------------------------------------------------------------
// 1) Build stable sort keys: (sortable(float) << 32) | edge_index
// ---------------------------------------------------------------------------
__global__ void fill_keys_kernel(const float* __restrict__ ew, int E,
                                 unsigned long long* __restrict__ keys) {
    int i = blockIdx.x * blockDim.x + threadIdx.x;
    if (i >= NPOW) return;
    if (i < E) {
        unsigned u = __float_as_uint(ew[i]);
        u ^= (u >> 31) ? 0xFFFFFFFFu : 0x80000000u;     // total order on floats
        keys[i] = ((unsigned long long)u << 32) | (unsigned)i;
    } else {
        keys[i] = ~0ull;                                 // padding sorts last
    }
}

// ---------------------------------------------------------------------------
// 2) Bitonic sort step (stable via index in low bits)
// ---------------------------------------------------------------------------
__global__ void bitonic_step_kernel(unsigned long long* __restrict__ keys, int j, int k) {
    int i = blockIdx.x * blockDim.x + threadIdx.x;
    if (i >= NPOW) return;
    int ixj = i ^ j;
    if (ixj <= i) return;
    unsigned long long a = keys[i], b = keys[ixj];
    bool up = ((i & k) == 0);
    if ((a > b) == up) { keys[i] = b; keys[ixj] = a; }
}

// ---------------------------------------------------------------------------
// 3) Gather edges into SORTED order as packed 16B records, so the serial
//    Kruskal thread streams one contiguous b128 per edge instead of three
//    random L2 loads on its dependent chain.
// ---------------------------------------------------------------------------
__global__ void gather_edges_kernel(const unsigned long long* __restrict__ keys,
                                    const int* __restrict__ src, const int* __restrict__ dst,
                                    const float* __restrict__ ew, int E,
                                    int4* __restrict__ edata) {
    int e = blockIdx.x * blockDim.x + threadIdx.x;
    if (e >= E) return;
    int ei = (int)(unsigned)(keys[e] & 0xFFFFFFFFull);
    edata[e] = make_int4(src[ei], dst[ei], __float_as_int(ew[ei]), ei);
}

// ---------------------------------------------------------------------------
// 4) Init tree arrays + output (out[n] = 0.5 for leaves: the "inside" leaf term)
// ---------------------------------------------------------------------------
__global__ void init_arrays_kernel(int* __restrict__ parent, float* __restrict__ sizes,
                                   float* __restrict__ alt, int* __restrict__ canon,
                                   float* __restrict__ out) {
    int i = blockIdx.x * blockDim.x + threadIdx.x;
    if (i >= NPOW) return;
    parent[i] = i;
    sizes[i]  = 1.0f;
    alt[i]    = 0.0f;
    if (i < Lleaves) canon[i] = i;
    if (i < N_NODES) out[i] = (i < Lleaves) ? 0.5f : 0.0f;
}

// ---------------------------------------------------------------------------
// 5) Serial Kruskal / BPT construction. The union-find array lives entirely
//    in the WGP's 320 KB LDS: entry >= 0 is a parent pointer; entry < 0 marks
//    a root with component size = -entry. Canonical node ids stay in a global
//    array (2 random loads + 1 store per merge). Semantics identical to the
//    reference (uf[a]=b link, same path compression, stable edge order).
// ---------------------------------------------------------------------------
__global__ void __launch_bounds__(1024)
kruskal_kernel(const int4* __restrict__ edata, int E,
               int* __restrict__ parent, float* __restrict__ sizes, float* __restrict__ alt,
               int* __restrict__ canon) {
    extern __shared__ int uf[];                       // Lleaves ints = 256 KB
    for (int i = threadIdx.x; i < Lleaves; i += blockDim.x) uf[i] = -1;  // root, size 1
    __syncthreads();
    if (threadIdx.x != 0) return;

    int node = Lleaves;
    for (int e = 0; e < E; ++e) {
        if ((e & 7) == 0) __builtin_prefetch(&edata[e + 32], 0, 0);  // global_prefetch_b8
        int4 rec = edata[e];
        int a = rec.x;
        { int r = a; while (uf[r] >= 0) r = uf[r];
          while (uf[a] >= 0) { int t = uf[a]; uf[a] = r; a = t; } a = r; }
        int b = rec.y;
        { int r = b; while (uf[r] >= 0) r = uf[r];
          while (uf[b] >= 0) { int t = uf[b]; uf[b] = r; b = t; } b = r; }
        if (a == b) continue;
        int sa = -uf[a], sb = -uf[b];
        int na = canon[a], nb = canon[b];
        parent[na] = node;
        parent[nb] = node;
        sizes[node] = (float)(sa + sb);
        alt[node]   = __int_as_float(rec.z);
        uf[a] = b;                                    // link (matches reference)
        uf[b] = -(sa + sb);                           // b stays root with merged size
        canon[b] = node;
        ++node;
    }
}

// ---------------------------------------------------------------------------
// 6) outside[n] = sum over ancestor path of (size[p]-size[cur]) * sig(alt[n]-alt[p])
//    Internal parent[] pointer-chased from the TDM-staged LDS copy.
// ---------------------------------------------------------------------------
__global__ void __launch_bounds__(512)
outside_kernel(const int* __restrict__ parent, const float* __restrict__ sizes,
               const float* __restrict__ alt, float* __restrict__ out) {
    extern __shared__ int sparent[];                  // NSTAGE ints = 256 KB
    stage_parents_to_lds(parent, sparent);
    for (int n = blockIdx.x * blockDim.x + threadIdx.x; n < N_NODES;
         n += blockDim.x * gridDim.x) {
        float altn = alt[n];
        float acc = 0.0f;
        float szcur = sizes[n];
        int cur = n;
        int p = (cur >= Lleaves) ? sparent[cur - Lleaves] : parent[cur];
        while (p != cur) {
            float szp = sizes[p];
            acc += (szp - szcur) * sigmoidf(altn - alt[p]);
            szcur = szp;
            cur = p;
            p = sparent[cur - Lleaves];               // all further nodes are internal
        }
        out[n] += acc;                                 // stream-ordered: no concurrency
    }
}

// ---------------------------------------------------------------------------
// 7) inside[a] += sig(alt[a]-alt[parent(x)]) for every leaf x under a.
//    One thread per leaf walks its ancestor chain (LDS-staged parents),
//    scattering with float atomics.
// ---------------------------------------------------------------------------
__global__ void __launch_bounds__(512)
inside_kernel(const int* __restrict__ parent, const float* __restrict__ alt,
              float* __restrict__ out) {
    extern __shared__ int sparent[];
    stage_parents_to_lds(parent, sparent);
    for (int x = blockIdx.x * blockDim.x + threadIdx.x; x < Lleaves;
         x += blockDim.x * gridDim.x) {
        int cur = parent[x];                           // grid is connected: cur != x
        float altp = alt[cur];
        while (true) {
            atomicAdd(&out[cur], sigmoidf(alt[cur] - altp));
            int nxt = sparent[cur - Lleaves];
            if (nxt == cur) break;
            cur = nxt;
        }
    }
}

// ---------------------------------------------------------------------------
extern "C" void kernel_launch(void* const* d_in, const int* in_sizes, int n_in,
                              void* d_out, int out_size, void* d_ws, size_t ws_size,
                              hipStream_t stream) {
    const float* ew  = (const float*)d_in[0];
    const int*   src = (const int*)d_in[1];
    const int*   dst = (const int*)d_in[2];
    float*       out = (float*)d_out;
    int E = in_sizes[0];

    // workspace layout (5 MB total)
    char* ws = (char*)d_ws;
    size_t off = 0;
    unsigned long long* keys = (unsigned long long*)(ws + off); off += (size_t)NPOW * 8;  // 1 MB
    int4*  edata  = (int4*) (ws + off); off += (size_t)NPOW * 16;                         // 2 MB
    int*   canon  = (int*)  (ws + off); off += (size_t)Lleaves * 4;                       // 256 KB
    int*   parent = (int*)  (ws + off); off += (size_t)NPOW * 4;                          // 512 KB
    float* sizes  = (float*)(ws + off); off += (size_t)NPOW * 4;                          // 512 KB
    float* alt    = (float*)(ws + off); off += (size_t)NPOW * 4;                          // 512 KB

    const int T = 256;
    const int B = NPOW / T;                            // 512 blocks

    fill_keys_kernel<<<B, T, 0, stream>>>(ew, E, keys);
    for (int k = 2; k <= NPOW; k <<= 1)
        for (int j = k >> 1; j > 0; j >>= 1)
            bitonic_step_kernel<<<B, T, 0, stream>>>(keys, j, k);

    gather_edges_kernel<<<(E + T - 1) / T, T, 0, stream>>>(keys, src, dst, ew, E, edata);
    init_arrays_kernel<<<B, T, 0, stream>>>(parent, sizes, alt, canon, out);

    kruskal_kernel<<<1, 1024, Lleaves * sizeof(int), stream>>>(
        edata, E, parent, sizes, alt, canon);

    outside_kernel<<<128, 512, NSTAGE * sizeof(int), stream>>>(parent, sizes, alt, out);
    inside_kernel <<<128, 512, NSTAGE * sizeof(int), stream>>>(parent, alt, out);
}